// Deu_30743375904929
// MI455X (gfx1250) — compile-verified
//
#include <hip/hip_runtime.h>
#include <math.h>

typedef __attribute__((ext_vector_type(4))) float v4f;

#define BLOCK            256
#define COLS_PER_THREAD  4
#define COLS_PER_BLOCK   (BLOCK * COLS_PER_THREAD)   // 1024 columns per block
#define ROWS_PER_BLOCK   16
#define PREFETCH_ROWS    4

// Per-column derived constants (registers for the whole row loop).
// e0,e1 are the (case-selected) transcendental argument multipliers.
// Real roots   (rr=1): f = c1*A + c2*E2 + sc*(1 + q1*A - q2*E2),  A=e^{r1 x}, E2=e^{r2 x}
// Complex roots(rr=0): f = (pcs-sc)*A*cos(b x) + sc + sc*aob*A*sin(b x),  A=e^{alpha x}
struct ColConst {
    float e0, e1;            // exp/trig argument multipliers (case-selected)
    float c1, c2, q1, q2;    // real-root path coefficients
    float pcs, aob;          // complex-root path coefficients
    float invc;
    bool  rr;                // true = real roots
};

// Branchless evaluation: compute both paths (4 trans total, shared args),
// combine with one select. No exec-mask divergence, full TRANS/VALU co-exec.
__device__ __forceinline__ float deu_eval(float xx, const ColConst& k) {
    float sc = (xx > 0.0f) ? k.invc : 0.0f;      // Heaviside/c folded

    float A  = __expf(k.e0 * xx);                // v_exp_f32 (shared)
    float t1 = k.e1 * xx;
    float E2 = __expf(t1);                       // v_exp_f32 (real path)
    float s  = __sinf(t1);                       // v_sin_f32 (complex path)
    float g  = __cosf(t1);                       // v_cos_f32 (complex path)

    // real-root result
    float u  = fmaf(k.q1, A, 1.0f);
    u        = fmaf(-k.q2, E2, u);
    float fr = k.c1 * A;
    fr       = fmaf(k.c2, E2, fr);
    fr       = fmaf(sc, u, fr);

    // complex-root result
    float fc = fmaf(k.pcs - sc, A * g, sc);
    fc       = fmaf(sc * k.aob, A * s, fc);

    return k.rr ? fr : fc;                       // v_cndmask_b32
}

__global__ __launch_bounds__(BLOCK) void deu_kernel(
    const float* __restrict__ x, const float* __restrict__ coeffs,
    float* __restrict__ out, int B, int N)
{
    const int tid  = threadIdx.x;
    const int col0 = blockIdx.x * COLS_PER_BLOCK + tid * COLS_PER_THREAD;
    const int row0 = blockIdx.y * ROWS_PER_BLOCK;
    if (col0 >= N) return;

    // ---- prologue: derive per-column constants once (amortized over rows) ----
    ColConst k[COLS_PER_THREAD];
    #pragma unroll
    for (int j = 0; j < COLS_PER_THREAD; ++j) {
        int col = col0 + j;
        if (col >= N) col = N - 1;               // clamp (tail safety)
        const float* cf = coeffs + (size_t)col * 5;
        float a = cf[0], b = cf[1], c = cf[2], c1 = cf[3], c2 = cf[4];

        float invc  = __builtin_amdgcn_rcpf(c);
        float inv2a = __builtin_amdgcn_rcpf(2.0f * a);
        float disc  = fmaf(b, b, -4.0f * a * c);
        float alpha = -b * inv2a;
        bool  rr    = (disc >= 0.0f);
        float w     = __builtin_amdgcn_sqrtf(fabsf(disc)) * inv2a; // delta | beta

        float rp    = alpha + w, rm = alpha - w;
        float inv2d = __builtin_amdgcn_rcpf(w + w);

        k[j].e0  = rr ? rp : alpha;              // only args need case-selection
        k[j].e1  = rr ? rm : w;
        k[j].c1  = c1;
        k[j].c2  = c2;
        k[j].q1  = rm * inv2d;                   // r2 / (2 delta)
        k[j].q2  = rp * inv2d;                   // r1 / (2 delta)
        k[j].pcs = c1 + c2;
        k[j].aob = (alpha * inv2d) * 2.0f;       // alpha / w
        k[j].invc = invc;
        k[j].rr  = rr;
    }

    const bool fullvec = (col0 + COLS_PER_THREAD <= N);

    // ---- hot loop: b128 NT load -> 4 branchless evals -> b128 NT store ----
    for (int r = 0; r < ROWS_PER_BLOCK; ++r) {
        int row = row0 + r;
        if (row >= B) break;
        size_t base = (size_t)row * (size_t)N + (size_t)col0;

        // speculative prefetch a few rows ahead (global_prefetch_b8; OOB dropped)
        __builtin_prefetch(x + base + (size_t)PREFETCH_ROWS * (size_t)N, 0, 1);

        if (fullvec) {
            v4f xv = __builtin_nontemporal_load((const v4f*)(x + base));
            v4f fv;
            #pragma unroll
            for (int j = 0; j < COLS_PER_THREAD; ++j)
                fv[j] = deu_eval(xv[j], k[j]);
            __builtin_nontemporal_store(fv, (v4f*)(out + base));
        } else {
            #pragma unroll
            for (int j = 0; j < COLS_PER_THREAD; ++j) {
                if (col0 + j < N) {
                    float xx = __builtin_nontemporal_load(x + base + j);
                    float f  = deu_eval(xx, k[j]);
                    __builtin_nontemporal_store(f, out + base + j);
                }
            }
        }
    }
}

extern "C" void kernel_launch(void* const* d_in, const int* in_sizes, int n_in,
                              void* d_out, int out_size, void* d_ws, size_t ws_size,
                              hipStream_t stream) {
    const float* x      = (const float*)d_in[0];
    const float* coeffs = (const float*)d_in[1];
    float*       out    = (float*)d_out;

    const int N = in_sizes[1] / 5;               // coeffs is [N,5]
    const int B = in_sizes[0] / N;               // x is [B,N]

    dim3 grid((N + COLS_PER_BLOCK - 1) / COLS_PER_BLOCK,
              (B + ROWS_PER_BLOCK - 1) / ROWS_PER_BLOCK);
    deu_kernel<<<grid, BLOCK, 0, stream>>>(x, coeffs, out, B, N);
}